// Net_25331717112263
// MI455X (gfx1250) — compile-verified
//
#include <hip/hip_runtime.h>
#include <math.h>

// ---------------- constants ----------------
#define N_NODES 50000
#define N_EDGES 400000
#define LOWER_CURV (-0.22703196f)
#define UPPER_CURV (0.36853024f)

typedef __attribute__((ext_vector_type(2))) float v2f;
typedef __attribute__((ext_vector_type(8))) float v8f;
typedef __attribute__((ext_vector_type(2))) int   v2i;

// Extended-K GEMM: B rows are [ W(125*CIN) | root(CIN) | bias(1) | zero-pad ]
template <int CIN> struct KDims {
    static constexpr int KT = 126 * CIN + 1;
    static constexpr int K4 = (KT + 3) / 4 * 4;
    static constexpr int CHUNKS = K4 / 4;
    static constexpr int STRIDE = K4 + 1;   // odd -> bank-conflict-free rows
};

// ---------------- small utility kernels ----------------
__global__ void zero_i32_k(int* p, int n) {
    int i = blockIdx.x * blockDim.x + threadIdx.x;
    if (i < n) p[i] = 0;
}

__global__ void copy_i32_k(const int* __restrict__ s, int* __restrict__ d, int n) {
    int i = blockIdx.x * blockDim.x + threadIdx.x;
    if (i < n) d[i] = s[i];
}

// degree-1 3D B-spline basis, kernel_size=5 per dim; packs (basis,wi) in 8 bytes
__global__ void basis_k(const float* __restrict__ attr, v2i* __restrict__ bw, int E) {
    int e = blockIdx.x * blockDim.x + threadIdx.x;
    if (e >= E) return;
    float f[3]; int k0[3];
#pragma unroll
    for (int d = 0; d < 3; d++) {
        float a = attr[e * 3 + d];
        a = fminf(fmaxf(a, 0.f), 1.f);
        float pos = a * 4.f;              // * (KSIZE-1)
        float fl = floorf(pos);
        if (fl > 3.f) fl = 3.f;           // clip to KSIZE-2
        k0[d] = (int)fl;
        f[d] = pos - fl;
    }
#pragma unroll
    for (int s = 0; s < 8; s++) {
        float b = 1.f; int id[3];
#pragma unroll
        for (int d = 0; d < 3; d++) {
            int bit = (s >> d) & 1;
            b *= bit ? f[d] : (1.f - f[d]);
            id[d] = k0[d] + bit;
        }
        v2i pk;
        pk.x = __float_as_int(b);
        pk.y = id[0] * 25 + id[1] * 5 + id[2];
        bw[e * 8 + s] = pk;
    }
}

__global__ void hist_k(const int* __restrict__ dst, int* __restrict__ deg, int E) {
    int e = blockIdx.x * blockDim.x + threadIdx.x;
    if (e < E) atomicAdd(&deg[dst[e]], 1);
}

// single-workgroup exclusive scan: offs[0]=0, offs[i+1]=sum(deg[0..i])
__global__ __launch_bounds__(1024) void scan_k(const int* __restrict__ deg,
                                               int* __restrict__ offs, int n) {
    __shared__ int buf[1024];
    __shared__ int carry;
    int t = threadIdx.x;
    if (t == 0) { carry = 0; offs[0] = 0; }
    __syncthreads();
    for (int base = 0; base < n; base += 1024) {
        int i = base + t;
        int v = (i < n) ? deg[i] : 0;
        buf[t] = v;
        __syncthreads();
        for (int o = 1; o < 1024; o <<= 1) {
            int tmp = (t >= o) ? buf[t - o] : 0;
            __syncthreads();
            buf[t] += tmp;
            __syncthreads();
        }
        int incl = buf[t] + carry;
        if (i < n) offs[i + 1] = incl;
        __syncthreads();
        if (t == 1023) carry = incl;
        __syncthreads();
    }
}

__global__ void scatter_k(const int* __restrict__ dst, int* __restrict__ cursor,
                          int* __restrict__ sortedE, int E) {
    int e = blockIdx.x * blockDim.x + threadIdx.x;
    if (e >= E) return;
    int p = atomicAdd(&cursor[dst[e]], 1);
    sortedE[p] = e;
}

// make CSR deterministic: sort each node's sublist by edge id (avg degree 8)
__global__ void sortlists_k(const int* __restrict__ offs, int* __restrict__ sortedE, int n) {
    int v = blockIdx.x * blockDim.x + threadIdx.x;
    if (v >= n) return;
    int st = offs[v], en = offs[v + 1];
    for (int a = st + 1; a < en; a++) {
        int key = sortedE[a];
        int b = a - 1;
        while (b >= st && sortedE[b] > key) { sortedE[b + 1] = sortedE[b]; b--; }
        sortedE[b + 1] = key;
    }
}

// permute per-edge data into dst-sorted order (hot loops then read sequentially)
__global__ void permute_k(const int* __restrict__ sortedE, const int* __restrict__ srcArr,
                          const v2i* __restrict__ bw, int* __restrict__ srcSorted,
                          v2i* __restrict__ bwS, int E) {
    int eIdx = blockIdx.x * blockDim.x + threadIdx.x;
    if (eIdx >= E) return;
    int e = sortedE[eIdx];
    srcSorted[eIdx] = srcArr[e];
#pragma unroll
    for (int s = 0; s < 8; s++) bwS[eIdx * 8 + s] = bw[e * 8 + s];
}

__global__ void transform_k(const float* __restrict__ x, float* __restrict__ h, int n) {
    int v = blockIdx.x * blockDim.x + threadIdx.x;
    if (v >= n) return;
    float t = (x[v] - LOWER_CURV) / (UPPER_CURV - LOWER_CURV) * 20.f - 10.f;
    t = fminf(fmaxf(t, -10.f), 10.f);
    h[v * 16] = t;
#pragma unroll
    for (int i = 1; i < 16; i++) h[v * 16 + i] = 0.f;
}

__global__ void copyout_k(const float* __restrict__ h, float* __restrict__ out, int n) {
    int i = blockIdx.x * blockDim.x + threadIdx.x;
    if (i < n) out[i] = h[i * 16];
}

// Pre-pack B operand in WMMA fragment order: for chunk c, lane l (hi=l>>4, m=l&15),
// Wlay[(c*32+l)*2 + j] = Brow[c*4 + 2*hi + j][m], with
// Brow[r][o] = W[r,o] (r<125*CIN) | root[r-125*CIN,o] | bias[o] (r==126*CIN) | 0.
template <int CIN, int COUT>
__device__ __forceinline__ float fetchB(const float* W, const float* root,
                                        const float* bias, int r, int o) {
    if (o >= COUT) return 0.f;
    if (r < 125 * CIN) return W[r * COUT + o];
    if (r < 126 * CIN) return root[(r - 125 * CIN) * COUT + o];
    if (r == 126 * CIN) return bias[o];
    return 0.f;
}

template <int CIN, int COUT>
__global__ void padw_k(const float* __restrict__ W, const float* __restrict__ root,
                       const float* __restrict__ bias, float* __restrict__ Wlay) {
    constexpr int CHUNKS = KDims<CIN>::CHUNKS;
    int t = blockIdx.x * blockDim.x + threadIdx.x;
    if (t >= CHUNKS * 32) return;
    int c = t >> 5, l = t & 31;
    int hi = l >> 4, m = l & 15;
    int kb = c * 4 + 2 * hi;
    Wlay[(size_t)t * 2 + 0] = fetchB<CIN, COUT>(W, root, bias, kb, m);
    Wlay[(size_t)t * 2 + 1] = fetchB<CIN, COUT>(W, root, bias, kb + 1, m);
}

// ---------------- fused spline-conv layer ----------------
// One workgroup (8 wave32s) owns 16 consecutive dst nodes.
// Phase B: accumulate h_tile[16][K] in LDS (lane-owned columns, deterministic,
//          no atomics) + identity columns for root/bias.
// Phase C: [16 x K] x [K x 16] GEMM via V_WMMA_F32_16X16X4_F32, K split over
//          8 waves; 2 chunks/iter with independent accumulators, depth-2 B
//          prefetch; LDS partial reduce, ELU, store.
template <int CIN, int COUT>
__global__ __launch_bounds__(256) void spline_layer_k(
    const float* __restrict__ hin, float* __restrict__ hout,
    const int* __restrict__ offs, const int* __restrict__ srcSorted,
    const v2i* __restrict__ bwS, const float* __restrict__ Wlay) {
    constexpr int CHUNKS = KDims<CIN>::CHUNKS;
    constexpr int STRIDE = KDims<CIN>::STRIDE;
    constexpr int SPAN = (32 / CIN) > 8 ? 8 : (32 / CIN);  // corners per pass

    __shared__ float hT[16 * STRIDE];
    __shared__ float partial[8 * 256];

    const int tid = threadIdx.x;
    const int tile = blockIdx.x;
    const int lane = tid & 31;
    const int wave = tid >> 5;
    const int m = lane & 15;   // A row / B col index
    const int hi = lane >> 4;  // lane half selects K pair

    const v2f* __restrict__ Bp = (const v2f*)Wlay;
    // Depth-2 B prefetch issued before Phase B: Wlay is shared by all 3125
    // blocks -> L2-resident; loads land while the edge loop runs.
    v2f b0 = Bp[wave * 32 + lane];
    v2f b1 = Bp[(size_t)((wave + 8 < CHUNKS) ? wave + 8 : wave) * 32 + lane];

    // ---- zero the LDS tile (padding columns stay 0) ----
    for (int i = tid; i < 16 * STRIDE; i += 256) hT[i] = 0.f;
    __syncthreads();

    // ---- stage root/bias identity columns (disjoint from edge columns) ----
    if (tid < 16 * CIN) {
        const int n = tid / CIN, i2 = tid % CIN;
        hT[n * STRIDE + 125 * CIN + i2] = hin[(tile * 16 + n) * 16 + i2];
    }
    if (tid < 16) hT[tid * STRIDE + 126 * CIN] = 1.f;

    // ---- Phase B: CSR edge accumulation, deterministic, no atomics ----
    {
        const int i = lane % CIN;            // input channel owned by lane
        const int sBase = lane / CIN;        // corner within a pass
        const bool active = lane < SPAN * CIN;
        for (int nn = 0; nn < 2; nn++) {
            const int n = wave * 2 + nn;     // 8 waves x 2 nodes = 16 nodes
            const int v = tile * 16 + n;
            const int st = offs[v], en = offs[v + 1];
            for (int eIdx = st; eIdx < en; eIdx++) {
                const int src = srcSorted[eIdx];
                const float xv = active ? hin[src * 16 + i] : 0.f;
#pragma unroll
                for (int p = 0; p < 8 / SPAN; p++) {
                    if (active) {
                        const v2i pk = bwS[(size_t)eIdx * 8 + sBase + p * SPAN];
                        // 8 corners of one edge have distinct wi, and each lane
                        // owns channel i -> column wi*CIN+i is lane-exclusive.
                        hT[n * STRIDE + pk.y * CIN + i] += __int_as_float(pk.x) * xv;
                    }
                }
            }
        }
    }
    __syncthreads();

    // ---- Phase C: WMMA GEMM  C[16x16] = A[16xK4] * B[K4x16] ----
    v8f acc0 = {}, acc1 = {};
    {
        // A (f32 16x4): lanes 0-15 M=0-15; VGPR0: K={0,2}, VGPR1: K={1,3}
        const float* __restrict__ hrow = &hT[m * STRIDE + 2 * hi];
        for (int c = wave; c < CHUNKS; c += 16) {
            const int cB = c + 8;                      // partner chunk
            const bool hasB = cB < CHUNKS;             // wave-uniform
            // prefetch B fragments one iteration ahead (clamped, branchless)
            const int cp0 = (c + 16 < CHUNKS) ? c + 16 : c;
            const int cp1 = (c + 24 < CHUNKS) ? c + 24 : c;
            const v2f bp0 = Bp[(size_t)cp0 * 32 + lane];
            const v2f bp1 = Bp[(size_t)cp1 * 32 + lane];
            // issue both A loads before either WMMA
            const int cBc = hasB ? cB : c;             // clamp A address
            v2f a0, a1;
            a0.x = hrow[c * 4];
            a0.y = hrow[c * 4 + 1];
            a1.x = hrow[cBc * 4];
            a1.y = hrow[cBc * 4 + 1];
            acc0 = __builtin_amdgcn_wmma_f32_16x16x4_f32(
                false, a0, false, b0, (short)0, acc0, false, false);
            if (hasB) {
                acc1 = __builtin_amdgcn_wmma_f32_16x16x4_f32(
                    false, a1, false, b1, (short)0, acc1, false, false);
            }
            b0 = bp0;
            b1 = bp1;
        }
    }
    acc0 += acc1;  // merge the two chains (fixed order -> deterministic)
    // stash per-wave partials: C element (vgpr j, lane) -> row j+8*hi, col m
#pragma unroll
    for (int j = 0; j < 8; j++) {
        partial[wave * 256 + (j + 8 * hi) * 16 + m] = acc0[j];
    }
    __syncthreads();

    // ---- reduce partials + ELU (root/bias already inside the GEMM) ----
    {
        const int M = tid >> 4;     // tile-local node row
        const int o = tid & 15;     // output channel
        float sum = 0.f;
#pragma unroll
        for (int w = 0; w < 8; w++) sum += partial[w * 256 + tid];
        float out = 0.f;
        if (o < COUT) out = sum > 0.f ? sum : (expf(sum) - 1.f);
        hout[(tile * 16 + M) * 16 + o] = out;
    }
}

// ---------------- host-side launch ----------------
extern "C" void kernel_launch(void* const* d_in, const int* in_sizes, int n_in,
                              void* d_out, int out_size, void* d_ws, size_t ws_size,
                              hipStream_t stream) {
    (void)in_sizes; (void)n_in; (void)out_size; (void)ws_size;
    const int N = N_NODES, E = N_EDGES;

    const float* x = (const float*)d_in[0];
    const int* ei = (const int*)d_in[1];
    const int* srcArr = ei;          // edge_index[0]
    const int* dstArr = ei + E;      // edge_index[1]
    const float* attr = (const float*)d_in[2];
    const float *W[5], *R[5], *B[5];
    for (int l = 0; l < 5; l++) {
        W[l] = (const float*)d_in[3 + 3 * l];
        R[l] = (const float*)d_in[4 + 3 * l];
        B[l] = (const float*)d_in[5 + 3 * l];
    }

    // carve workspace (~60 MB total)
    char* ws = (char*)d_ws;
    size_t off = 0;
    auto carve = [&](size_t bytes) -> void* {
        void* p = ws + off;
        off = (off + bytes + 255) & ~(size_t)255;
        return p;
    };
    v2i*   bw      = (v2i*)carve((size_t)E * 8 * 8);
    v2i*   bwS     = (v2i*)carve((size_t)E * 8 * 8);
    int*   deg     = (int*)carve((size_t)(N + 1) * 4);
    int*   offs    = (int*)carve((size_t)(N + 1) * 4);
    int*   cursor  = (int*)carve((size_t)N * 4);
    int*   sortedE = (int*)carve((size_t)E * 4);
    int*   srcSorted = (int*)carve((size_t)E * 4);
    float* hA      = (float*)carve((size_t)N * 16 * 4);
    float* hB      = (float*)carve((size_t)N * 16 * 4);
    // per-layer packed B operands: 16*K4 floats each (K4: 128,1012,2020,2020,1012)
    const int K4s[5] = {128, 1012, 2020, 2020, 1012};
    float* Wlay[5];
    for (int l = 0; l < 5; l++) Wlay[l] = (float*)carve((size_t)16 * K4s[l] * 4);

    const int TB = 256;
    // one-time graph preprocessing (basis + deterministic CSR-by-dst + permute)
    basis_k<<<(E + TB - 1) / TB, TB, 0, stream>>>(attr, bw, E);
    zero_i32_k<<<(N + 1 + TB - 1) / TB, TB, 0, stream>>>(deg, N + 1);
    hist_k<<<(E + TB - 1) / TB, TB, 0, stream>>>(dstArr, deg, E);
    scan_k<<<1, 1024, 0, stream>>>(deg, offs, N);
    copy_i32_k<<<(N + TB - 1) / TB, TB, 0, stream>>>(offs, cursor, N);
    scatter_k<<<(E + TB - 1) / TB, TB, 0, stream>>>(dstArr, cursor, sortedE, E);
    sortlists_k<<<(N + TB - 1) / TB, TB, 0, stream>>>(offs, sortedE, N);
    permute_k<<<(E + TB - 1) / TB, TB, 0, stream>>>(sortedE, srcArr, bw, srcSorted, bwS, E);

    // pack B operands (W | root | bias, fragment order)
    padw_k<1, 8><<<(K4s[0] / 4 * 32 + TB - 1) / TB, TB, 0, stream>>>(W[0], R[0], B[0], Wlay[0]);
    padw_k<8, 16><<<(K4s[1] / 4 * 32 + TB - 1) / TB, TB, 0, stream>>>(W[1], R[1], B[1], Wlay[1]);
    padw_k<16, 16><<<(K4s[2] / 4 * 32 + TB - 1) / TB, TB, 0, stream>>>(W[2], R[2], B[2], Wlay[2]);
    padw_k<16, 8><<<(K4s[3] / 4 * 32 + TB - 1) / TB, TB, 0, stream>>>(W[3], R[3], B[3], Wlay[3]);
    padw_k<8, 1><<<(K4s[4] / 4 * 32 + TB - 1) / TB, TB, 0, stream>>>(W[4], R[4], B[4], Wlay[4]);

    transform_k<<<(N + TB - 1) / TB, TB, 0, stream>>>(x, hA, N);

    const int TILES = N / 16;  // 50000 = 3125 * 16 exactly
    spline_layer_k<1, 8><<<TILES, 256, 0, stream>>>(hA, hB, offs, srcSorted, bwS, Wlay[0]);
    spline_layer_k<8, 16><<<TILES, 256, 0, stream>>>(hB, hA, offs, srcSorted, bwS, Wlay[1]);
    spline_layer_k<16, 16><<<TILES, 256, 0, stream>>>(hA, hB, offs, srcSorted, bwS, Wlay[2]);
    spline_layer_k<16, 8><<<TILES, 256, 0, stream>>>(hB, hA, offs, srcSorted, bwS, Wlay[3]);
    spline_layer_k<8, 1><<<TILES, 256, 0, stream>>>(hA, hB, offs, srcSorted, bwS, Wlay[4]);

    copyout_k<<<(N + TB - 1) / TB, TB, 0, stream>>>(hB, (float*)d_out, N);
}